// SepConv_79912161509631
// MI455X (gfx1250) — compile-verified
//
#include <hip/hip_runtime.h>

// Problem constants (fixed by the reference)
#define HH   181
#define WW   360
#define CIN  32
#define COUT 64
#define KK   4
#define NNZ  9
#define NNZ1 4
#define HW   (HH * WW)            // 65160
#define NP_TILES ((HW + 15) / 16) // 4073 column tiles of 16

typedef __attribute__((ext_vector_type(2))) float v2f;
typedef __attribute__((ext_vector_type(8))) float v8f;

// ---------------------------------------------------------------------------
// Kernel A: xq = x*quad_w ; xd = disco(xq) ; y[c,h,w] = sum_k w_depth[c,k]*xd
// Entries for segment (k,h) are contiguous: e = (k*H + h)*9 + j.
// Grid = (ceil(W/256), H, CIN) so vals/hi/wi addresses are wave-uniform.
// ---------------------------------------------------------------------------
__global__ void disco_depth_kernel(const float* __restrict__ x,
                                   const float* __restrict__ quad_w,
                                   const float* __restrict__ vals,
                                   const int*   __restrict__ hi,
                                   const int*   __restrict__ wi,
                                   const float* __restrict__ w_depth,
                                   float*       __restrict__ y) {
    const int w = blockIdx.x * blockDim.x + threadIdx.x;
    const int h = blockIdx.y;
    const int c = blockIdx.z;
    if (w >= WW) return;

    float acc = 0.0f;
#pragma unroll
    for (int k = 0; k < KK; ++k) {
        const float wd    = w_depth[c * KK + k];
        const int   ebase = (k * HH + h) * NNZ;
#pragma unroll
        for (int j = 0; j < NNZ; ++j) {
            const int e   = ebase + j;
            const int hie = hi[e];
            int col = w + wi[e];              // wi in {0, 1, W-1}
            if (col >= WW) col -= WW;         // circular shift
            acc += wd * vals[e] * quad_w[hie] * x[c * HW + hie * WW + col];
        }
    }
    y[c * HW + h * WW + w] = acc;
}

// ---------------------------------------------------------------------------
// Kernel B: yq = y*quad_w ; z = disco1(yq)   (4 entries per row: e = h*4 + j)
// ---------------------------------------------------------------------------
__global__ void disco_second_kernel(const float* __restrict__ y,
                                    const float* __restrict__ quad_w,
                                    const float* __restrict__ vals1,
                                    const int*   __restrict__ hi1,
                                    const int*   __restrict__ wi1,
                                    float*       __restrict__ z) {
    const int w = blockIdx.x * blockDim.x + threadIdx.x;
    const int h = blockIdx.y;
    const int c = blockIdx.z;
    if (w >= WW) return;

    float acc = 0.0f;
#pragma unroll
    for (int j = 0; j < NNZ1; ++j) {
        const int e   = h * NNZ1 + j;
        const int hie = hi1[e];
        int col = w + wi1[e];
        if (col >= WW) col -= WW;
        acc += vals1[e] * quad_w[hie] * y[c * HW + hie * WW + col];
    }
    z[c * HW + h * WW + w] = acc;
}

// ---------------------------------------------------------------------------
// Kernel C: out[o,p] = sum_c w_point[o,c]*z[c,p] + bias[o]
//   GEMM  M=COUT=64, K=CIN=32, N=HW=65160  via V_WMMA_F32_16X16X4_F32.
//
// Per ISA VGPR layouts (wave32):
//   A 16x4 f32 : lane L (m = L&15):  VGPR0 = A[m, K=2*(L>>4)],  VGPR1 = +1
//   B 4x16 f32 : lane L (n = L&15):  VGPR0 = B[K=2*(L>>4), n],  VGPR1 = +1
//   C/D 16x16  : VGPR r, lane L: element (M = r + 8*(L>>4), N = L&15)
//
// One wave: loads the K=32 B-panel once (8 chunks x v2f) for one 16-wide
// column tile, then accumulates 8 chained WMMAs for each of the 4 M-tiles
// of w_point (32 WMMAs / wave). EXEC stays all-ones across WMMA: edge tiles
// clamp load addresses and predicate only the stores.
// ---------------------------------------------------------------------------
__global__ void pointwise_wmma_kernel(const float* __restrict__ z,
                                      const float* __restrict__ w_point,
                                      const float* __restrict__ bias,
                                      float*       __restrict__ out) {
    const int lane  = threadIdx.x & 31;
    const int wave  = threadIdx.x >> 5;
    const int ptile = blockIdx.x * 8 + wave;       // 8 waves per block
    const bool live = (ptile < NP_TILES);

    const int p0   = (live ? ptile : 0) * 16;
    const int half = lane >> 4;                    // 0: lanes 0-15, 1: 16-31
    const int l16  = lane & 15;
    const int koff = half * 2;                     // K offset inside 4-chunk

    const int pcol   = p0 + l16;
    const int pcol_c = (pcol < HW) ? pcol : (HW - 1);   // clamped edge load

    // Load full B-panel (32 x 16 tile of z): 8 k-chunks of v2f
    v2f bmat[8];
#pragma unroll
    for (int kc = 0; kc < 8; ++kc) {
        const int krow = kc * 4 + koff;
        bmat[kc].x = z[(krow + 0) * HW + pcol_c];
        bmat[kc].y = z[(krow + 1) * HW + pcol_c];
    }

#pragma unroll
    for (int ob = 0; ob < 4; ++ob) {               // 4 M-tiles of 16 rows
        const int orow = ob * 16 + l16;            // A row for this lane
        v8f acc = {};
#pragma unroll
        for (int kc = 0; kc < 8; ++kc) {           // K = 32 in 8 x4 steps
            v2f a;
            a.x = w_point[orow * CIN + kc * 4 + koff + 0];
            a.y = w_point[orow * CIN + kc * 4 + koff + 1];
            // D = A x B + C   (8 args: neg_a, A, neg_b, B, c_mod, C, reuse_a, reuse_b)
            acc = __builtin_amdgcn_wmma_f32_16x16x4_f32(
                false, a, false, bmat[kc], (short)0, acc, false, false);
        }
        // Scatter D + bias (predicated stores only; WMMA already done)
#pragma unroll
        for (int r = 0; r < 8; ++r) {
            const int o = ob * 16 + r + 8 * half;
            if (live && pcol < HW) {
                out[o * HW + pcol] = acc[r] + bias[o];
            }
        }
    }
}

// ---------------------------------------------------------------------------
// Launcher.  setup_inputs() order:
//  0:x 1:quad_w 2:vals 3:seg 4:hi 5:wi 6:w_depth
//  7:vals1 8:seg1 9:hi1 10:wi1 11:w_point 12:bias 13:K
// seg/seg1/K unused (segment indices are directly computable).
// Workspace: y (HW*CIN f32) then z (HW*CIN f32) = 16.7 MB total.
// ---------------------------------------------------------------------------
extern "C" void kernel_launch(void* const* d_in, const int* in_sizes, int n_in,
                              void* d_out, int out_size, void* d_ws, size_t ws_size,
                              hipStream_t stream) {
    const float* x       = (const float*)d_in[0];
    const float* quad_w  = (const float*)d_in[1];
    const float* vals    = (const float*)d_in[2];
    const int*   hi      = (const int*)  d_in[4];
    const int*   wi      = (const int*)  d_in[5];
    const float* w_depth = (const float*)d_in[6];
    const float* vals1   = (const float*)d_in[7];
    const int*   hi1     = (const int*)  d_in[9];
    const int*   wi1     = (const int*)  d_in[10];
    const float* w_point = (const float*)d_in[11];
    const float* bias    = (const float*)d_in[12];
    float*       out     = (float*)d_out;

    float* y = (float*)d_ws;
    float* z = y + (size_t)CIN * HW;

    dim3 blk(256);
    dim3 grdA((WW + 255) / 256, HH, CIN);          // (2, 181, 32)
    disco_depth_kernel<<<grdA, blk, 0, stream>>>(x, quad_w, vals, hi, wi,
                                                 w_depth, y);

    disco_second_kernel<<<grdA, blk, 0, stream>>>(y, quad_w, vals1, hi1, wi1, z);

    dim3 grdC((NP_TILES + 7) / 8);                 // 8 waves (tiles) per block
    pointwise_wmma_kernel<<<grdC, blk, 0, stream>>>(z, w_point, bias, out);
}